// Model_91036126806535
// MI455X (gfx1250) — compile-verified
//
#include <hip/hip_runtime.h>

typedef float v2f __attribute__((ext_vector_type(2)));
typedef float v8f __attribute__((ext_vector_type(8)));

#define MDIM  64   // feature dim (reference M)
#define CHUNK 64   // rows per workgroup

__global__ void zero_out_kernel(float* __restrict__ out, int n) {
  int i = blockIdx.x * blockDim.x + threadIdx.x;
  if (i < n) out[i] = 0.0f;
}

// first index in off[0..n) with off[idx] > v  (searchsorted 'right')
__device__ __forceinline__ int upper_bound_dev(const int* __restrict__ off, int n, int v) {
  int lo = 0, hi = n;
  while (lo < hi) {
    int mid = (lo + hi) >> 1;
    if (off[mid] <= v) lo = mid + 1; else hi = mid;
  }
  return lo;
}

// One workgroup = 128 threads = 4 wave32's. Each block owns 64 consecutive
// rows of `values`; wave w owns columns [16w, 16w+16). Segment-sum of the
// chunk is D = S x V via chained V_WMMA_F32_16X16X4_F32:
//   S (16 local-segments x 4 rows) = 0/1 selection matrix  (A operand)
//   V (4 rows x 16 cols)           = values tile           (B operand)
// A operands are pre-materialized from LDS, all 32 B loads are issued up
// front (non-temporal), so the WMMA chain runs back-to-back with only
// descending s_wait_loadcnt. Cross-chunk merge: sparse global_atomic_add_f32.
__global__ __launch_bounds__(128)
void jagged_sum_wmma(const float* __restrict__ values,
                     const int*   __restrict__ offsets,
                     float*       __restrict__ out,
                     int total, int n_off) {
  __shared__ int s_seg[CHUNK];   // absolute segment id per row
  __shared__ int s_ls[CHUNK];    // segment id local to this chunk

  const int r0 = blockIdx.x * CHUNK;
  const int t  = threadIdx.x;

  // segment id for each of the 64 rows in this chunk (monotone)
  if (t < CHUNK) {
    int row = r0 + t;
    if (row >= total) row = total - 1;
    s_seg[t] = upper_bound_dev(offsets, n_off, row) - 1;
  }
  __syncthreads();

  const int seg_lo = s_seg[0];
  const int span   = s_seg[CHUNK - 1] - seg_lo;     // uniform across block
  const bool full  = (r0 + CHUNK) <= total;         // uniform across block

  // localize once so per-lane selection is a bare compare against lane id
  if (t < CHUNK) s_ls[t] = s_seg[t] - seg_lo;
  __syncthreads();

  if (full && span <= 15) {
    // ---- WMMA fast path (taken for ~all chunks: avg segment ~1024 rows) ----
    const int lane = t & 31;
    const int wave = t >> 5;
    const int half = lane >> 4;   // 0: lanes 0-15, 1: lanes 16-31
    const int mn   = lane & 15;   // A: local-segment row M; B/D: column N
    const int c0   = wave * 16;   // this wave's 16-column tile

    // Pre-materialize A: lane holds A[M=mn][K = half*2 + {0,1}] per k-block.
    v2f a[16];
#pragma unroll
    for (int i = 0; i < 16; ++i) {
      const int k0 = i * 4 + half * 2;
      a[i].x = (s_ls[k0]     == mn) ? 1.0f : 0.0f;
      a[i].y = (s_ls[k0 + 1] == mn) ? 1.0f : 0.0f;
    }

    // Issue all B-operand loads up front: values streamed exactly once,
    // non-temporal (256MB total > 192MB L2, single use).
    const float* base = values + (size_t)(r0 + half * 2) * MDIM + c0 + mn;
    v2f b[16];
#pragma unroll
    for (int i = 0; i < 16; ++i) {
      b[i].x = __builtin_nontemporal_load(base + (size_t)(4 * i) * MDIM);
      b[i].y = __builtin_nontemporal_load(base + (size_t)(4 * i + 1) * MDIM);
    }

    v8f acc = {0.f, 0.f, 0.f, 0.f, 0.f, 0.f, 0.f, 0.f};
#pragma unroll
    for (int i = 0; i < 16; ++i) {
      acc = __builtin_amdgcn_wmma_f32_16x16x4_f32(
          /*neg_a=*/false, a[i], /*neg_b=*/false, b[i],
          /*c_mod=*/(short)0, acc, /*reuse_a=*/false, /*reuse_b=*/false);
    }

    // Flush only local segments that occur (ls <= span): sparse f32 atomics
    // merge partials across chunk boundaries (~16-32 atomics per wave).
#pragma unroll
    for (int v = 0; v < 8; ++v) {
      const int ls = v + half * 8;          // D row M held by this lane/VGPR
      if (ls <= span) {
        atomicAdd(&out[(size_t)(seg_lo + ls) * MDIM + c0 + mn], acc[v]);
      }
    }
  } else {
    // ---- rare fallback: tail chunk or >16 segments in this chunk ----
    if (t < MDIM) {
      for (int k = 0; k < CHUNK; ++k) {
        const int row = r0 + k;
        if (row < total) {
          atomicAdd(&out[(size_t)s_seg[k] * MDIM + t],
                    values[(size_t)row * MDIM + t]);
        }
      }
    }
  }
}

extern "C" void kernel_launch(void* const* d_in, const int* in_sizes, int n_in,
                              void* d_out, int out_size, void* d_ws, size_t ws_size,
                              hipStream_t stream) {
  const float* values  = (const float*)d_in[0];
  const int*   offsets = (const int*)d_in[1];
  float*       out     = (float*)d_out;

  const int total = in_sizes[0] / MDIM;   // jagged row count
  const int n_off = in_sizes[1];          // B + 1

  // d_out is poisoned by the harness; atomically accumulated -> zero it first.
  zero_out_kernel<<<(out_size + 255) / 256, 256, 0, stream>>>(out, out_size);

  const int nblocks = (total + CHUNK - 1) / CHUNK;
  jagged_sum_wmma<<<nblocks, 128, 0, stream>>>(values, offsets, out, total, n_off);
}